// BiRWKVAttention_11579231830777
// MI455X (gfx1250) — compile-verified
//
#include <hip/hip_runtime.h>
#include <hip/hip_bf16.h>

// BiRWKV on MI455X (gfx1250): 3-pass blocked scan, fp32 WMMA for intra-chunk
// triangular prefix. Bandwidth-bound (~268MB min HBM traffic ~ 12us floor at
// 23.3 TB/s). k+v (134MB) stay resident in the 192MB L2 between pass 1 and
// pass 3; streaming-once data (r, out, carries) uses non-temporal hints so it
// does not evict that working set.

#define B_SZ   4
#define T_LEN  4096
#define C_DIM  1024
#define CHUNK  16
#define NCH    (T_LEN / CHUNK)   // 256

typedef __attribute__((ext_vector_type(2))) float v2f;
typedef __attribute__((ext_vector_type(8))) float v8f;

// ---------------------------------------------------------------------------
// Pass 1: per-(b, chunk, c) aggregates.
//   fa = sum_j ew^(15-j) * e^k_j * v_j     (forward numerator aggregate)
//   fb = sum_j ew^(15-j) * e^k_j           (forward denominator aggregate)
//   ba = sum_j ew^j      * e^k_j * v_j     (backward numerator aggregate)
//   bb = sum_j ew^j      * e^k_j           (backward denominator aggregate)
// 1M threads; coalesced along c (innermost dim). k,v read with default (RT)
// policy on purpose: pass 3 re-reads them and they fit in L2.
// ---------------------------------------------------------------------------
__global__ __launch_bounds__(256) void k_chunk_agg(
    const float* __restrict__ k_in, const float* __restrict__ v_in,
    const float* __restrict__ w_in,
    float* __restrict__ fa, float* __restrict__ fb,
    float* __restrict__ ba, float* __restrict__ bb) {
  int id = blockIdx.x * 256 + threadIdx.x;        // (b*NCH + n)*C + c
  int c    = id & (C_DIM - 1);
  int rest = id >> 10;
  int n    = rest & (NCH - 1);
  int b    = rest >> 8;

  float ww = w_in[c];
  float ew = __expf(ww);

  size_t g = ((size_t)(b * T_LEN + n * CHUNK)) * C_DIM + c;
  float af = 0.f, bf = 0.f, ab = 0.f, db = 0.f, p = 1.f;
#pragma unroll
  for (int j = 0; j < CHUNK; ++j) {
    float kk  = k_in[g + (size_t)j * C_DIM];
    float vv  = v_in[g + (size_t)j * C_DIM];
    float ek  = __expf(kk);
    float ekv = ek * vv;
    af = ew * af + ekv;       // ends as sum ew^(15-j) ekv_j
    bf = ew * bf + ek;
    ab += p * ekv;            // sum ew^j ekv_j
    db += p * ek;
    p  *= ew;
  }
  fa[id] = af; fb[id] = bf; ba[id] = ab; bb[id] = db;
}

// ---------------------------------------------------------------------------
// Pass 2: scan chunk aggregates -> chunk-entry carries, IN PLACE.
// After this kernel:
//   fa[b][n][c] = carryF_a(n) = sum_{i < n*16} ew^(n*16-1-i) ekv_i
//   ba[b][n][c] = carryB_a(n) = sum_{i >= (n+1)*16} ew^(i-(n+1)*16) ekv_i
// (fb/bb analogous with e^k only.) 4096 threads, 32MB of scratch traffic.
// ---------------------------------------------------------------------------
__global__ __launch_bounds__(256) void k_carry_scan(
    const float* __restrict__ w_in,
    float* __restrict__ fa, float* __restrict__ fb,
    float* __restrict__ ba, float* __restrict__ bb) {
  int id = blockIdx.x * 256 + threadIdx.x;  // b*C + c
  int c = id & (C_DIM - 1);
  int b = id >> 10;
  float ew16 = __expf((float)CHUNK * w_in[c]);
  size_t base = (size_t)b * NCH * C_DIM + c;

  float cfa = 0.f, cfb = 0.f;
#pragma unroll 4
  for (int n = 0; n < NCH; ++n) {
    size_t idx = base + (size_t)n * C_DIM;
    float ta = fa[idx], tb = fb[idx];
    fa[idx] = cfa; fb[idx] = cfb;           // carry entering chunk n
    cfa = ew16 * cfa + ta;                  // absorb chunk n
    cfb = ew16 * cfb + tb;
  }
  float cba = 0.f, cbb = 0.f;
#pragma unroll 4
  for (int n = NCH - 1; n >= 0; --n) {
    size_t idx = base + (size_t)n * C_DIM;
    float ta = ba[idx], tb = bb[idx];
    ba[idx] = cba; bb[idx] = cbb;           // carry entering chunk n from right
    cba = ew16 * cba + ta;
    cbb = ew16 * cbb + tb;
  }
}

// ---------------------------------------------------------------------------
// Pass 3: intra-chunk strict triangular prefix via fp32 WMMA + final combine.
// One wave handles a 16(time) x 16(channel) tile of one chunk.
//   Pf[tau][c] = ew^tau      * (L_strict @ Xf)[tau][c],  Xf[j] = ew^(-(j+1)) x_j
//   Pb[tau][c] = ew^(-tau-1) * (U_strict @ Xb)[tau][c],  Xb[j] = ew^j        x_j
// Each of the 4 products (fwd/bwd x num/den) = 4 chained v_wmma_f32_16x16x4_f32.
// ---------------------------------------------------------------------------
__global__ __launch_bounds__(256) void k_wkv_wmma(
    const float* __restrict__ r_in, const float* __restrict__ k_in,
    const float* __restrict__ v_in, const float* __restrict__ w_in,
    const float* __restrict__ u_in,
    const float* __restrict__ cfa, const float* __restrict__ cfb,
    const float* __restrict__ cba, const float* __restrict__ cbb,
    float* __restrict__ out) {
  __shared__ float lds[8][4][CHUNK][17];  // 8 waves x {Xfa,Xfb,Xba,Xbb} x 16x16(+pad)

  int wave = threadIdx.x >> 5;
  int lane = threadIdx.x & 31;
  int lo   = lane & 15;       // N index (channel within tile) / M index for A
  int hi   = lane >> 4;       // half-wave selector

  int cpb  = C_DIM / 128;                        // channel blocks per (b,n): 8
  int cblk = blockIdx.x % cpb;
  int n    = (blockIdx.x / cpb) % NCH;
  int b    =  blockIdx.x / (cpb * NCH);
  int c    = cblk * 128 + wave * 16 + lo;
  int t0   = n * CHUNK;

  float ww = w_in[c];
  size_t gbase = ((size_t)(b * T_LEN + t0)) * C_DIM + c;

  float (*Xfa)[17] = lds[wave][0];
  float (*Xfb)[17] = lds[wave][1];
  float (*Xba)[17] = lds[wave][2];
  float (*Xbb)[17] = lds[wave][3];

  // Load tile, build rescaled operands. Lane covers (t = 2i+hi, channel lo).
  // k,v read with default policy (expected L2-resident from pass 1).
#pragma unroll
  for (int i = 0; i < 8; ++i) {
    int t = 2 * i + hi;
    float kk  = k_in[gbase + (size_t)t * C_DIM];
    float vv  = v_in[gbase + (size_t)t * C_DIM];
    float ek  = __expf(kk);
    float ekv = ek * vv;
    float pF  = __expf(-ww * (float)(t + 1));   // ew^(-(t+1))
    float pB  = __expf( ww * (float)t);         // ew^t
    Xfa[t][lo] = pF * ekv;  Xfb[t][lo] = pF * ek;
    Xba[t][lo] = pB * ekv;  Xbb[t][lo] = pB * ek;
  }
  __syncthreads();

  // K=16 triangular matmuls as 4 chained 16x16x4 f32 WMMAs each.
  // A layout (16x4 f32): lane holds A[M=lo][K = r + 2*hi], r in {0,1}.
  // B layout (4x16 f32): lane holds B[K = r + 2*hi][N = lo].
  // D layout (16x16 f32): lane holds D[M = r + 8*hi][N = lo], r in 0..7.
  v8f Dfa = {}, Dfb = {}, Dba = {}, Dbb = {};
#pragma unroll
  for (int q = 0; q < 4; ++q) {
    int j0 = 4 * q + 2 * hi;  // the two K rows this lane supplies
    v2f aF = { (j0     < lo) ? 1.f : 0.f, (j0 + 1 < lo) ? 1.f : 0.f };  // strict lower
    v2f aB = { (j0     > lo) ? 1.f : 0.f, (j0 + 1 > lo) ? 1.f : 0.f };  // strict upper
    v2f bfa = { Xfa[j0][lo], Xfa[j0 + 1][lo] };
    v2f bfb = { Xfb[j0][lo], Xfb[j0 + 1][lo] };
    v2f bba = { Xba[j0][lo], Xba[j0 + 1][lo] };
    v2f bbb = { Xbb[j0][lo], Xbb[j0 + 1][lo] };
    Dfa = __builtin_amdgcn_wmma_f32_16x16x4_f32(false, aF, false, bfa, (short)0, Dfa, false, false);
    Dfb = __builtin_amdgcn_wmma_f32_16x16x4_f32(false, aF, false, bfb, (short)0, Dfb, false, false);
    Dba = __builtin_amdgcn_wmma_f32_16x16x4_f32(false, aB, false, bba, (short)0, Dba, false, false);
    Dbb = __builtin_amdgcn_wmma_f32_16x16x4_f32(false, aB, false, bbb, (short)0, Dbb, false, false);
  }

  // Combine with carries and elementwise tail. Carries and r are read-once,
  // out is write-once: non-temporal so they don't evict k/v from L2.
  size_t cidx = ((size_t)(b * NCH + n)) * C_DIM + c;
  float vfa = __builtin_nontemporal_load(&cfa[cidx]);
  float vfb = __builtin_nontemporal_load(&cfb[cidx]);
  float vba = __builtin_nontemporal_load(&cba[cidx]);
  float vbb = __builtin_nontemporal_load(&cbb[cidx]);
  float uu  = u_in[c];

#pragma unroll
  for (int rr = 0; rr < 8; ++rr) {
    int   tau = rr + 8 * hi;
    size_t g  = gbase + (size_t)tau * C_DIM;
    float ewt = __expf( ww * (float)tau);                 // ew^tau
    float ewm = __expf(-ww * (float)(tau + 1));           // ew^(-(tau+1))
    float ewr = __expf( ww * (float)(CHUNK - 1 - tau));   // ew^(15-tau)

    float Af = ewt * (vfa + Dfa[rr]);
    float Bf = ewt * (vfb + Dfb[rr]);
    float Ab = ewr * vba + ewm * Dba[rr];
    float Bb = ewr * vbb + ewm * Dbb[rr];

    float kk = k_in[g], vv = v_in[g];
    float rv = __builtin_nontemporal_load(&r_in[g]);
    float cur = __expf(uu + kk);
    float wkv = (Af + Ab + cur * vv) / (Bf + Bb + cur);
    __builtin_nontemporal_store(wkv / (1.f + __expf(-rv)), &out[g]);
  }
}

// ---------------------------------------------------------------------------
extern "C" void kernel_launch(void* const* d_in, const int* in_sizes, int n_in,
                              void* d_out, int out_size, void* d_ws, size_t ws_size,
                              hipStream_t stream) {
  (void)in_sizes; (void)n_in; (void)out_size; (void)ws_size;
  const float* r = (const float*)d_in[0];
  const float* k = (const float*)d_in[1];
  const float* v = (const float*)d_in[2];
  const float* w = (const float*)d_in[3];
  const float* u = (const float*)d_in[4];
  float* out = (float*)d_out;

  float* ws = (float*)d_ws;                     // needs 4 * B*NCH*C floats = 16 MB
  size_t sz = (size_t)B_SZ * NCH * C_DIM;
  float* fa = ws;           float* fb = ws + sz;
  float* ba = ws + 2 * sz;  float* bb = ws + 3 * sz;

  k_chunk_agg <<<(B_SZ * NCH * C_DIM) / 256, 256, 0, stream>>>(k, v, w, fa, fb, ba, bb);
  k_carry_scan<<<(B_SZ * C_DIM) / 256,       256, 0, stream>>>(w, fa, fb, ba, bb);
  k_wkv_wmma  <<<B_SZ * NCH * (C_DIM / 128), 256, 0, stream>>>(r, k, v, w, u,
                                                               fa, fb, ba, bb, out);
}